// E74CheckpointedCell_90520730730970
// MI455X (gfx1250) — compile-verified
//
#include <hip/hip_runtime.h>
#include <math.h>

// Problem constants (from reference): T=512, B=32, DIM=1024, N=256
#define T_DIM 512
#define B_DIM 32
#define D_DIM 1024
#define N_DIM 256
#define M_DIM (T_DIM * B_DIM)   // 16384 rows for the projection GEMMs

// Scan-kernel chunking / LDS layout
#define C_DIM 16                // chunk length (our choice; math is exact)
#define SSTR  258               // S row stride in floats (bank-conflict free, 8B aligned)
#define KSTR  258               // K/Q/U chunk-row stride
#define GSTR  36                // 16x16 + pad for A/G

typedef __attribute__((ext_vector_type(2))) float v2f;
typedef __attribute__((ext_vector_type(8))) float v8f;
typedef __attribute__((ext_vector_type(4))) int   v4i;

static __device__ __forceinline__ v8f wmma_f32(v2f a, v2f b, v8f c) {
    // D = A(16x4) * B(4x16) + C(16x16), all fp32  ->  v_wmma_f32_16x16x4_f32
    return __builtin_amdgcn_wmma_f32_16x16x4_f32(false, a, false, b,
                                                 (short)0, c, false, false);
}

// ---------------------------------------------------------------------------
// Async global -> LDS staging (GLOBAL_LOAD_ASYNC_TO_LDS_B128, ASYNCcnt).
// Builtin signature (from hipcc diagnostic): param0 = AS1 (global) int4*,
// param1 = LDS dest, then imm offset + cpol. Guarded fallback keeps the
// synchronous VGPR round-trip if the builtin is ever absent.
// ---------------------------------------------------------------------------
#if defined(__has_builtin)
#  if __has_builtin(__builtin_amdgcn_global_load_async_to_lds_b128)
#    define HAVE_ASYNC_B128 1
#  endif
#  if __has_builtin(__builtin_amdgcn_s_wait_asynccnt)
#    define HAVE_WAIT_ASYNC 1
#  endif
#endif
#ifndef HAVE_ASYNC_B128
#define HAVE_ASYNC_B128 0
#endif
#ifndef HAVE_WAIT_ASYNC
#define HAVE_WAIT_ASYNC 0
#endif

typedef __attribute__((address_space(1))) v4i* gv4i_p;   // global int4*
typedef __attribute__((address_space(3))) v4i* lv4i_p;   // LDS int4*

static __device__ __forceinline__ void stage16(void* lds_dst, const void* gsrc) {
#if HAVE_ASYNC_B128
    void* gnc = const_cast<void*>(gsrc);
    __builtin_amdgcn_global_load_async_to_lds_b128(
        (gv4i_p)gnc, (lv4i_p)lds_dst, 0, 0);
#else
    *(float4*)lds_dst = *(const float4*)gsrc;
#endif
}

static __device__ __forceinline__ void stage_wait() {
#if HAVE_ASYNC_B128
#  if HAVE_WAIT_ASYNC
    __builtin_amdgcn_s_wait_asynccnt(0);
#  else
    asm volatile("s_wait_asynccnt 0x0" ::: "memory");
#  endif
#endif
}

// ---------------------------------------------------------------------------
// Kernel 1: fused projections  Y = X * W^T  for W_k, W_v, W_q  (fp32 WMMA)
// ---------------------------------------------------------------------------
__global__ __launch_bounds__(32)
void proj_kernel(const float* __restrict__ x,
                 const float* __restrict__ Wk,
                 const float* __restrict__ Wv,
                 const float* __restrict__ Wq,
                 float* __restrict__ kout,
                 float* __restrict__ vout,
                 float* __restrict__ qout)
{
    __shared__ float Xs[16][36];
    __shared__ float Ws[3][16][36];

    const int lane = (int)threadIdx.x;
    const int m0   = (int)blockIdx.x * 16;
    const int n0   = (int)blockIdx.y * 16;
    const int r    = lane >> 1;
    const int c    = (lane & 1) * 16;

    v8f ak = {}, av = {}, aq = {};

    for (int d0 = 0; d0 < D_DIM; d0 += 32) {
        {
            const float* src = x + (size_t)(m0 + r) * D_DIM + d0 + c;
            float* dst = &Xs[r][c];
#pragma unroll
            for (int u = 0; u < 4; ++u) stage16(dst + 4 * u, src + 4 * u);
        }
#define LOAD_W(idx, Wptr) { \
        const float* src = (Wptr) + (size_t)(n0 + r) * D_DIM + d0 + c; \
        float* dst = &Ws[idx][r][c]; \
        _Pragma("unroll") \
        for (int u = 0; u < 4; ++u) stage16(dst + 4 * u, src + 4 * u); }
        LOAD_W(0, Wk)
        LOAD_W(1, Wv)
        LOAD_W(2, Wq)
#undef LOAD_W
        stage_wait();
        __syncthreads();

        const int mi = lane & 15;
        const int kh = (lane >> 4) * 2;
#pragma unroll
        for (int kk = 0; kk < 32; kk += 4) {
            v2f a;  a.x  = Xs[mi][kk + kh];     a.y  = Xs[mi][kk + kh + 1];
            v2f bk; bk.x = Ws[0][mi][kk + kh];  bk.y = Ws[0][mi][kk + kh + 1];
            v2f bv; bv.x = Ws[1][mi][kk + kh];  bv.y = Ws[1][mi][kk + kh + 1];
            v2f bq; bq.x = Ws[2][mi][kk + kh];  bq.y = Ws[2][mi][kk + kh + 1];
            ak = wmma_f32(a, bk, ak);
            av = wmma_f32(a, bv, av);
            aq = wmma_f32(a, bq, aq);
        }
        __syncthreads();
    }

    const int hi = lane >> 4;
    const int nn = lane & 15;
#pragma unroll
    for (int rr = 0; rr < 8; ++rr) {
        const size_t off = (size_t)(m0 + rr + 8 * hi) * N_DIM + (size_t)(n0 + nn);
        kout[off] = ak[rr];
        vout[off] = av[rr];
        qout[off] = aq[rr];
    }
}

// ---------------------------------------------------------------------------
// Kernel 2: in-place row L2-normalization of k:  k /= (||k||_2 + 1e-6)
// ---------------------------------------------------------------------------
__global__ __launch_bounds__(256)
void norm_kernel(float* __restrict__ k)
{
    const int wave = (int)threadIdx.x >> 5;
    const int lane = (int)threadIdx.x & 31;
    const int row  = (int)blockIdx.x * 8 + wave;

    float* p = k + (size_t)row * N_DIM + lane * 8;
    float4 f0 = *(const float4*)(p + 0);
    float4 f1 = *(const float4*)(p + 4);
    float s = f0.x*f0.x + f0.y*f0.y + f0.z*f0.z + f0.w*f0.w
            + f1.x*f1.x + f1.y*f1.y + f1.z*f1.z + f1.w*f1.w;
#pragma unroll
    for (int off = 16; off > 0; off >>= 1) s += __shfl_down(s, off, 32);
    s = __shfl(s, 0, 32);
    const float inv = 1.0f / (sqrtf(s) + 1e-6f);
    f0.x *= inv; f0.y *= inv; f0.z *= inv; f0.w *= inv;
    f1.x *= inv; f1.y *= inv; f1.z *= inv; f1.w *= inv;
    *(float4*)(p + 0) = f0;
    *(float4*)(p + 4) = f1;
}

// ---------------------------------------------------------------------------
// Kernel 3: chunked WY delta-rule scan, fp32 WMMA end-to-end.
// One workgroup (8 waves) per batch; S[b] resident in LDS the whole time.
// Per chunk of C=16 steps:
//   R = K S0^T, P = Q S0^T        (GEMMs, shared S^T B-frags)
//   A = K K^T (wave0), G = Q K^T (wave1)
//   U = (I + tril(A,-1))^{-1} (V - R)   (16 sequential micro-steps, col-parallel)
//   O = P + tril(G) U  -> O^2*sigmoid(O) -> out   (accumulated into live P accumulators)
//   S += U^T K                          (256 16x16 tiles of WMMA)
// ---------------------------------------------------------------------------
__global__ __launch_bounds__(256)
void scan_wmma_kernel(const float* __restrict__ kb,
                      const float* __restrict__ vb,
                      const float* __restrict__ qb,
                      const float* __restrict__ S0,
                      float* __restrict__ out)
{
    extern __shared__ float smem[];
    float* Ss = smem;                               // N_DIM * SSTR
    float* Ks = Ss + (size_t)N_DIM * SSTR;          // C_DIM * KSTR
    float* Qs = Ks + C_DIM * KSTR;                  // C_DIM * KSTR
    float* Us = Qs + C_DIM * KSTR;                  // C_DIM * KSTR (V -> V-R -> U)
    float* As = Us + C_DIM * KSTR;                  // C_DIM * GSTR
    float* Gs = As + C_DIM * GSTR;                  // C_DIM * GSTR

    const int b    = (int)blockIdx.x;
    const int tid  = (int)threadIdx.x;
    const int wave = tid >> 5;
    const int lane = tid & 31;
    const int lo16 = lane & 15;
    const int hi   = lane >> 4;
    const int kh   = hi * 2;

    // ---- load S0[b] into LDS (async, coalesced) ----
    {
        const int jc = (tid & 15) * 16;
        const int rb = tid >> 4;
#pragma unroll
        for (int ib = 0; ib < 16; ++ib) {
            const int i = ib * 16 + rb;
            const float* g = S0 + ((size_t)b * N_DIM + i) * N_DIM + jc;
            float* dst = Ss + (size_t)i * SSTR + jc;
#pragma unroll
            for (int u = 0; u < 4; ++u) stage16(dst + 4 * u, g + 4 * u);
        }
        stage_wait();
    }

    const int i0a = wave * 32;        // this wave's two output column tiles
    const int i0b = wave * 32 + 16;

    for (int c = 0; c < T_DIM / C_DIM; ++c) {
        const int t0 = c * C_DIM;
        __syncthreads();   // prev chunk's S update done; chunk buffers free

        // ---- stage K, Q, V rows for this chunk (async, coalesced) ----
        {
            const int t  = tid >> 4;
            const int jc = (tid & 15) * 16;
            const size_t gbase = ((size_t)(t0 + t) * B_DIM + b) * N_DIM + jc;
            const float* gk = kb + gbase;
            const float* gq = qb + gbase;
            const float* gv = vb + gbase;
            float* dk = Ks + (size_t)t * KSTR + jc;
            float* dq = Qs + (size_t)t * KSTR + jc;
            float* dv = Us + (size_t)t * KSTR + jc;
#pragma unroll
            for (int u = 0; u < 4; ++u) {
                stage16(dk + 4 * u, gk + 4 * u);
                stage16(dq + 4 * u, gq + 4 * u);
                stage16(dv + 4 * u, gv + 4 * u);
            }
            stage_wait();
        }
        __syncthreads();

        // ---- R = K S^T and P = Q S^T (shared S^T fragments) ----
        v8f accR0 = {}, accR1 = {}, accP0 = {}, accP1 = {};
        for (int d = 0; d < N_DIM; d += 4) {
            v2f aK; aK.x = Ks[lo16 * KSTR + d + kh]; aK.y = Ks[lo16 * KSTR + d + kh + 1];
            v2f aQ; aQ.x = Qs[lo16 * KSTR + d + kh]; aQ.y = Qs[lo16 * KSTR + d + kh + 1];
            v2f b0; b0.x = Ss[(size_t)(i0a + lo16) * SSTR + d + kh];
                    b0.y = Ss[(size_t)(i0a + lo16) * SSTR + d + kh + 1];
            v2f b1; b1.x = Ss[(size_t)(i0b + lo16) * SSTR + d + kh];
                    b1.y = Ss[(size_t)(i0b + lo16) * SSTR + d + kh + 1];
            accR0 = wmma_f32(aK, b0, accR0);
            accP0 = wmma_f32(aQ, b0, accP0);
            accR1 = wmma_f32(aK, b1, accR1);
            accP1 = wmma_f32(aQ, b1, accP1);
        }
        // VR = V - R (in Us); disjoint (t, i) per lane across waves
#pragma unroll
        for (int r = 0; r < 8; ++r) {
            const int t = r + 8 * hi;
            Us[(size_t)t * KSTR + i0a + lo16] -= accR0[r];
            Us[(size_t)t * KSTR + i0b + lo16] -= accR1[r];
        }

        // ---- A = K K^T (wave 0), G = Q K^T masked to tril (wave 1) ----
        if (wave == 0) {
            v8f acc = {};
            for (int d = 0; d < N_DIM; d += 4) {
                v2f a; a.x = Ks[lo16 * KSTR + d + kh]; a.y = Ks[lo16 * KSTR + d + kh + 1];
                acc = wmma_f32(a, a, acc);   // A and B frags carry identical lane data
            }
#pragma unroll
            for (int r = 0; r < 8; ++r)
                As[(r + 8 * hi) * GSTR + lo16] = acc[r];
        } else if (wave == 1) {
            v8f acc = {};
            for (int d = 0; d < N_DIM; d += 4) {
                v2f a;  a.x  = Qs[lo16 * KSTR + d + kh]; a.y  = Qs[lo16 * KSTR + d + kh + 1];
                v2f bk; bk.x = Ks[lo16 * KSTR + d + kh]; bk.y = Ks[lo16 * KSTR + d + kh + 1];
                acc = wmma_f32(a, bk, acc);
            }
#pragma unroll
            for (int r = 0; r < 8; ++r) {
                const int t = r + 8 * hi;
                const int s = lo16;
                Gs[t * GSTR + s] = (s <= t) ? acc[r] : 0.0f;
            }
        }
        __syncthreads();

        // ---- forward substitution: U = (I + tril(A,-1))^{-1} (V - R), in place ----
        {
            const int j = tid;           // this thread owns column j
            float u[C_DIM];
            u[0] = Us[0 * KSTR + j];
#pragma unroll
            for (int t = 1; t < C_DIM; ++t) {
                float acc = Us[t * KSTR + j];
#pragma unroll
                for (int s = 0; s < t; ++s) acc -= As[t * GSTR + s] * u[s];
                u[t] = acc;
                Us[t * KSTR + j] = acc;
            }
        }
        __syncthreads();

        // ---- O = P + tril(G) U, then out = O^2 * sigmoid(O) ----
#pragma unroll
        for (int kk = 0; kk < C_DIM; kk += 4) {
            v2f a;  a.x  = Gs[lo16 * GSTR + kk + kh]; a.y  = Gs[lo16 * GSTR + kk + kh + 1];
            v2f b0; b0.x = Us[(kk + kh) * KSTR + i0a + lo16];
                    b0.y = Us[(kk + kh + 1) * KSTR + i0a + lo16];
            v2f b1; b1.x = Us[(kk + kh) * KSTR + i0b + lo16];
                    b1.y = Us[(kk + kh + 1) * KSTR + i0b + lo16];
            accP0 = wmma_f32(a, b0, accP0);
            accP1 = wmma_f32(a, b1, accP1);
        }
#pragma unroll
        for (int r = 0; r < 8; ++r) {
            const int m = r + 8 * hi;
            const size_t ob = ((size_t)(t0 + m) * B_DIM + b) * N_DIM;
            const float o0 = accP0[r];
            const float o1 = accP1[r];
            out[ob + i0a + lo16] = o0 * o0 / (1.0f + __expf(-o0));
            out[ob + i0b + lo16] = o1 * o1 / (1.0f + __expf(-o1));
        }

        // ---- S += U^T K : 256 tiles of 16x16, K'=16 ----
        for (int tile = wave; tile < 256; tile += 8) {
            const int ti = (tile >> 4) * 16;   // i block (rows of S)
            const int tj = (tile & 15) * 16;   // j block (cols of S)
            v8f cacc;
#pragma unroll
            for (int r = 0; r < 8; ++r)
                cacc[r] = Ss[(size_t)(ti + r + 8 * hi) * SSTR + tj + lo16];
#pragma unroll
            for (int kk = 0; kk < C_DIM; kk += 4) {
                v2f a;  a.x  = Us[(kk + kh) * KSTR + ti + lo16];
                        a.y  = Us[(kk + kh + 1) * KSTR + ti + lo16];
                v2f bb; bb.x = Ks[(kk + kh) * KSTR + tj + lo16];
                        bb.y = Ks[(kk + kh + 1) * KSTR + tj + lo16];
                cacc = wmma_f32(a, bb, cacc);
            }
#pragma unroll
            for (int r = 0; r < 8; ++r)
                Ss[(size_t)(ti + r + 8 * hi) * SSTR + tj + lo16] = cacc[r];
        }
    }
}

// ---------------------------------------------------------------------------
extern "C" void kernel_launch(void* const* d_in, const int* in_sizes, int n_in,
                              void* d_out, int out_size, void* d_ws, size_t ws_size,
                              hipStream_t stream)
{
    (void)in_sizes; (void)n_in; (void)out_size; (void)ws_size;

    const float* x  = (const float*)d_in[0];   // [T,B,DIM]
    const float* S0 = (const float*)d_in[1];   // [B,N,N]
    const float* Wk = (const float*)d_in[2];   // [N,DIM]
    const float* Wv = (const float*)d_in[3];   // [N,DIM]
    const float* Wq = (const float*)d_in[4];   // [N,DIM]
    float* out = (float*)d_out;                // [T,B,N]

    float* kbuf = (float*)d_ws;                          // M_DIM*N_DIM floats
    float* vbuf = kbuf + (size_t)M_DIM * N_DIM;
    float* qbuf = vbuf + (size_t)M_DIM * N_DIM;

    // 1) projections (fp32 WMMA)
    dim3 pgrid(M_DIM / 16, N_DIM / 16);
    proj_kernel<<<pgrid, 32, 0, stream>>>(x, Wk, Wv, Wq, kbuf, vbuf, qbuf);

    // 2) k row-normalization (in place)
    norm_kernel<<<M_DIM / 8, 256, 0, stream>>>(kbuf);

    // 3) chunked WMMA scan, one workgroup per batch, S resident in LDS
    const size_t lds_bytes =
        ((size_t)N_DIM * SSTR + 3 * C_DIM * KSTR + 2 * C_DIM * GSTR) * sizeof(float);
    (void)hipFuncSetAttribute(reinterpret_cast<const void*>(scan_wmma_kernel),
                              hipFuncAttributeMaxDynamicSharedMemorySize,
                              (int)lds_bytes);
    scan_wmma_kernel<<<B_DIM, 256, lds_bytes, stream>>>(kbuf, vbuf, qbuf, S0, out);
}